// GCN_88390426952000
// MI455X (gfx1250) — compile-verified
//
#include <hip/hip_runtime.h>
#include <hip/hip_bf16.h>

typedef float v2f __attribute__((ext_vector_type(2)));
typedef float v4f __attribute__((ext_vector_type(4)));
typedef float v8f __attribute__((ext_vector_type(8)));
typedef int   async_v4i __attribute__((vector_size(16)));   // matches builtin proto

// ---------------------------------------------------------------------------
// Degree accumulation (float atomics -> global_atomic_add_f32)
// ---------------------------------------------------------------------------
__global__ void gcn_edge_deg(const int* __restrict__ src, const int* __restrict__ dst,
                             float* __restrict__ deg_out, float* __restrict__ deg_in, int E) {
    int e = blockIdx.x * blockDim.x + threadIdx.x;
    if (e < E) {
        unsafeAtomicAdd(&deg_out[src[e]], 1.0f);
        unsafeAtomicAdd(&deg_in[dst[e]], 1.0f);
    }
}

// deg -> rsqrt(max(deg,1)); both norm arrays are contiguous, so n == 2*N
__global__ void gcn_norm(float* __restrict__ d, int n) {
    int i = blockIdx.x * blockDim.x + threadIdx.x;
    if (i < n) {
        float v = d[i];
        v = v < 1.0f ? 1.0f : v;
        d[i] = rsqrtf(v);
    }
}

// ---------------------------------------------------------------------------
// Y[N,Mout] = diag(rowscale) * (X[N,K] @ W[K,Mout])  via V_WMMA_F32_16X16X4_F32
// block = 256 threads (8 waves). Block owns 16 rows; wave w owns cols
// [blockIdx.y*128 + w*16, +16). A-tile staged into LDS with CDNA5 async
// global->LDS DMA (GLOBAL_LOAD_ASYNC_TO_LDS_B128, ASYNCcnt); row-scaling
// folded into the C-store epilogue so the staging is a pure copy.
// LDS row stride = K+4 floats: keeps 16B alignment and is bank-conflict-free
// (stride mod 64 banks = 4 -> 16 distinct banks across lanes 0-15).
// ---------------------------------------------------------------------------
__global__ __launch_bounds__(256)
void gcn_gemm_wmma(const float* __restrict__ X, const float* __restrict__ rowscale,
                   const float* __restrict__ W, float* __restrict__ Y,
                   int K, int Mout) {
    extern __shared__ float As[];              // 16 x (K+4)
    const int row0   = blockIdx.x * 16;
    const int tid    = threadIdx.x;
    const int stride = K + 4;
    const int kvecs  = K >> 2;                 // float4 chunks per row

    for (int i = tid; i < 16 * kvecs; i += 256) {
        int r  = i / kvecs;
        int c4 = (i - r * kvecs) << 2;
        const float* gsrc = X + (size_t)(row0 + r) * K + c4;
        float*       ldst = As + r * stride + c4;
#if __has_builtin(__builtin_amdgcn_global_load_async_to_lds_b128)
        __builtin_amdgcn_global_load_async_to_lds_b128(
            (__attribute__((address_space(1))) async_v4i*)gsrc,
            (__attribute__((address_space(3))) async_v4i*)ldst,
            /*offset=*/0, /*cpol=*/0);
#else
        *(v4f*)ldst = *(const v4f*)gsrc;
#endif
    }
#if __has_builtin(__builtin_amdgcn_s_wait_asynccnt)
    __builtin_amdgcn_s_wait_asynccnt(0);
#else
    asm volatile("s_wait_asynccnt 0x0" ::: "memory");
#endif
    __syncthreads();

    const int wave = tid >> 5;
    const int lane = tid & 31;
    const int half = lane >> 4;                // 0: lanes 0-15, 1: lanes 16-31
    const int l16  = lane & 15;
    const int cg   = blockIdx.y * 128 + wave * 16 + l16;   // global column
    const int kp   = half * 2;                 // K sub-offset per ISA A/B layout

    v8f acc = {};
    for (int k = 0; k < K; k += 4) {
        v2f a, b;
        a.x = As[l16 * stride + k + kp];
        a.y = As[l16 * stride + k + kp + 1];
        b.x = W[(size_t)(k + kp)     * Mout + cg];
        b.y = W[(size_t)(k + kp + 1) * Mout + cg];
        acc = __builtin_amdgcn_wmma_f32_16x16x4_f32(
            /*neg_a=*/false, a, /*neg_b=*/false, b,
            /*c_mod=*/(short)0, acc, /*reuse_a=*/false, /*reuse_b=*/false);
    }

    #pragma unroll
    for (int v = 0; v < 8; ++v) {
        int row = row0 + v + half * 8;         // C layout: lanes16-31 hold M=v+8
        Y[(size_t)row * Mout + cg] = acc[v] * rowscale[row];
    }
}

// ---------------------------------------------------------------------------
// agg[dst[e], :] += H[src[e], :]  — one thread per (edge, channel)
// ---------------------------------------------------------------------------
__global__ void gcn_spmm(const float* __restrict__ H, const int* __restrict__ src,
                         const int* __restrict__ dst, float* __restrict__ agg,
                         int E, int logC) {
    long long t = (long long)blockIdx.x * blockDim.x + threadIdx.x;
    int e = (int)(t >> logC);
    int c = (int)(t & ((1 << logC) - 1));
    if (e < E) {
        int s = src[e];
        int d = dst[e];
        unsafeAtomicAdd(&agg[((size_t)d << logC) + c], H[((size_t)s << logC) + c]);
    }
}

// H = relu(agg * norm_dst + b)
__global__ void gcn_relu_bias(const float* __restrict__ agg, const float* __restrict__ ndst,
                              const float* __restrict__ bias, float* __restrict__ H,
                              long long total, int logC) {
    long long t = (long long)blockIdx.x * blockDim.x + threadIdx.x;
    if (t < total) {
        int i = (int)(t >> logC);
        int c = (int)(t & ((1 << logC) - 1));
        float v = agg[t] * ndst[i] + bias[c];
        H[t] = v > 0.0f ? v : 0.0f;
    }
}

// ---------------------------------------------------------------------------
// out[N,4] = H[N,256] @ Wc[256,4] + bc — one wave32 per node
// ---------------------------------------------------------------------------
__global__ __launch_bounds__(256)
void gcn_classifier(const float* __restrict__ H, const float* __restrict__ Wc,
                    const float* __restrict__ bc, float* __restrict__ out, int N) {
    int gt   = blockIdx.x * blockDim.x + threadIdx.x;
    int node = gt >> 5;
    int lane = gt & 31;
    if (node >= N) return;

    float s0 = 0.f, s1 = 0.f, s2 = 0.f, s3 = 0.f;
    const float* hrow = H + (size_t)node * 256;
    for (int k = lane; k < 256; k += 32) {
        float h = hrow[k];
        s0 += h * Wc[k * 4 + 0];
        s1 += h * Wc[k * 4 + 1];
        s2 += h * Wc[k * 4 + 2];
        s3 += h * Wc[k * 4 + 3];
    }
    #pragma unroll
    for (int off = 16; off > 0; off >>= 1) {
        s0 += __shfl_down(s0, off);
        s1 += __shfl_down(s1, off);
        s2 += __shfl_down(s2, off);
        s3 += __shfl_down(s3, off);
    }
    if (lane == 0) {
        out[(size_t)node * 4 + 0] = s0 + bc[0];
        out[(size_t)node * 4 + 1] = s1 + bc[1];
        out[(size_t)node * 4 + 2] = s2 + bc[2];
        out[(size_t)node * 4 + 3] = s3 + bc[3];
    }
}

// ---------------------------------------------------------------------------
extern "C" void kernel_launch(void* const* d_in, const int* in_sizes, int n_in,
                              void* d_out, int out_size, void* d_ws, size_t ws_size,
                              hipStream_t stream) {
    const float* x   = (const float*)d_in[0];
    const int*   src = (const int*)  d_in[1];
    const int*   dst = (const int*)  d_in[2];
    const float* W0  = (const float*)d_in[3];
    const float* b0  = (const float*)d_in[4];
    const float* W1  = (const float*)d_in[5];
    const float* b1  = (const float*)d_in[6];
    const float* W2  = (const float*)d_in[7];
    const float* b2  = (const float*)d_in[8];
    const float* W3  = (const float*)d_in[9];
    const float* b3  = (const float*)d_in[10];
    const float* Wc  = (const float*)d_in[11];
    const float* bc  = (const float*)d_in[12];
    float* out = (float*)d_out;

    const int N = in_sizes[0] / 128;   // 100000 (divisible by 16)
    const int E = in_sizes[1];         // 800000

    // workspace carve-up
    float* norm_src = (float*)d_ws;                 // N
    float* norm_dst = norm_src + N;                 // N  (contiguous with norm_src)
    float* bufA = norm_dst + N;                     // N*256  (GEMM output)
    float* bufB = bufA + (size_t)N * 256;           // N*256  (aggregation)
    float* bufC = bufB + (size_t)N * 256;           // N*256  (activations)

    // ---- degree norms ----
    (void)hipMemsetAsync(norm_src, 0, (size_t)2 * N * sizeof(float), stream);
    gcn_edge_deg<<<(E + 255) / 256, 256, 0, stream>>>(src, dst, norm_src, norm_dst, E);
    gcn_norm<<<(2 * N + 255) / 256, 256, 0, stream>>>(norm_src, 2 * N);

    // ---- one GCN layer ----
    auto layer = [&](const float* Xin, int K, int C, const float* W, const float* b) {
        dim3 grid(N / 16, C / 128);
        size_t lds = (size_t)16 * (K + 4) * sizeof(float);
        gcn_gemm_wmma<<<grid, 256, lds, stream>>>(Xin, norm_src, W, bufA, K, C);

        (void)hipMemsetAsync(bufB, 0, (size_t)N * C * sizeof(float), stream);
        int logC = (C == 128) ? 7 : 8;
        long long etot = (long long)E << logC;
        gcn_spmm<<<(unsigned)((etot + 255) / 256), 256, 0, stream>>>(bufA, src, dst, bufB, E, logC);

        long long ntot = (long long)N << logC;
        gcn_relu_bias<<<(unsigned)((ntot + 255) / 256), 256, 0, stream>>>(bufB, norm_dst, b, bufC, ntot, logC);
    };

    layer(x,    128, 128, W0, b0);
    layer(bufC, 128, 128, W1, b1);
    layer(bufC, 128, 256, W2, b2);
    layer(bufC, 256, 256, W3, b3);

    gcn_classifier<<<(N * 32 + 255) / 256, 256, 0, stream>>>(bufC, Wc, bc, out, N);
}